// VQC_73375221285529
// MI455X (gfx1250) — compile-verified
//
#include <hip/hip_runtime.h>

// ---------------------------------------------------------------------------
// 16-qubit, 10-layer VQC state-vector simulation, batch 128.
// State (64 MB complex64) lives in d_ws (L2-resident on MI455X, 192MB L2).
// Per layer:
//   k_cross : CNOT(15,0) + R(wire0)    (only ops crossing the bit-15 split)
//   k_local : R(1..15) fused with next layer's CNOTs, on a 256KB LDS tile
//             (320KB LDS per WGP on CDNA5 makes the half-state tile fit)
// Fused 2-wire gates are 8x8 real matrices applied with V_WMMA_F32_16X16X4_F32.
// Tile global<->LDS movement uses gfx1250 async-to-LDS ops (ASYNCcnt).
// ---------------------------------------------------------------------------

constexpr int NW   = 16;
constexpr int NL   = 10;
constexpr int AMPS = 1 << NW;    // 65536 amplitudes per batch element
constexpr int HALF = AMPS >> 1;  // 32768 amps per tile (bit15 fixed)
constexpr int PPB  = 3 * NW * NL; // 480 params per batch element
constexpr int PPL  = 3 * NW;      // 48 params per layer

typedef float v2f __attribute__((ext_vector_type(2)));
typedef float v8f __attribute__((ext_vector_type(8)));
typedef int   v4i __attribute__((ext_vector_type(4)));

#define VQC_AS1 __attribute__((address_space(1)))
#define VQC_AS3 __attribute__((address_space(3)))

#if defined(__gfx1250__) && __has_builtin(__builtin_amdgcn_wmma_f32_16x16x4_f32)
#define VQC_HAVE_WMMA 1
#endif

#if defined(__gfx1250__) && \
    __has_builtin(__builtin_amdgcn_global_load_async_to_lds_b128) && \
    __has_builtin(__builtin_amdgcn_s_wait_asynccnt)
#define VQC_HAVE_ASYNC_LD 1
#endif

#if defined(__gfx1250__) && \
    __has_builtin(__builtin_amdgcn_global_store_async_from_lds_b128) && \
    __has_builtin(__builtin_amdgcn_s_wait_asynccnt)
#define VQC_HAVE_ASYNC_ST 1
#endif

struct cplx { float re, im; };
__device__ __forceinline__ cplx cmul(cplx a, cplx b) {
  return { a.re * b.re - a.im * b.im, a.re * b.im + a.im * b.re };
}
__device__ __forceinline__ cplx cadd(cplx a, cplx b) {
  return { a.re + b.re, a.im + b.im };
}

// U = RZ(tz) * RY(ty) * RX(tx)  (matches reference _rot_u)
__device__ void rot_u(float tx, float ty, float tz, cplx U[2][2]) {
  float cx = cosf(0.5f * tx), sx = sinf(0.5f * tx);
  float cy = cosf(0.5f * ty), sy = sinf(0.5f * ty);
  float cz = cosf(0.5f * tz), sz = sinf(0.5f * tz);
  // T = RY * RX
  cplx t00 = { cy * cx,  sy * sx };
  cplx t01 = { -sy * cx, -cy * sx };
  cplx t10 = { sy * cx,  -cy * sx };
  cplx t11 = { cy * cx,  -sy * sx };
  cplx ez  = { cz, -sz };   // e^{-i tz/2}
  cplx ezc = { cz,  sz };   // e^{+i tz/2}
  U[0][0] = cmul(ez, t00);  U[0][1] = cmul(ez, t01);
  U[1][0] = cmul(ezc, t10); U[1][1] = cmul(ezc, t11);
}

// real-ification of a complex entry g into the (er, ec) slot of a 2x2 block
// [[Re,-Im],[Im,Re]]
__device__ __forceinline__ float creal_entry(cplx g, int er, int ec) {
  return ec ? (er ? g.re : -g.im) : (er ? g.im : g.re);
}

#ifdef VQC_HAVE_WMMA
__device__ __forceinline__ v8f wmma4(v2f a, v2f b, v8f c) {
  // D(16x16) = A(16x4) x B(4x16) + C
  return __builtin_amdgcn_wmma_f32_16x16x4_f32(false, a, false, b,
                                               (short)0, c, false, false);
}
#endif

// ---------------------------------------------------------------------------
// k_init: state = |0...0> for each batch element.
// ---------------------------------------------------------------------------
__global__ void vqc_init(float2* __restrict__ state) {
  int idx = blockIdx.x * blockDim.x + threadIdx.x;  // 0 .. 128*65536-1
  float2 v;
  v.x = ((idx & (AMPS - 1)) == 0) ? 1.0f : 0.0f;
  v.y = 0.0f;
  state[idx] = v;
}

// ---------------------------------------------------------------------------
// k_cross: CNOT(15,0) then R(wire0) for layer `layer`.
// wire15 = bit0 (control), wire0 = bit15 (target / rotated).
// ---------------------------------------------------------------------------
__global__ void vqc_cross(const float* __restrict__ params,
                          float2* __restrict__ state, int layer) {
  int b = blockIdx.y;
  int i = blockIdx.x * blockDim.x + threadIdx.x;  // 0 .. HALF-1 (bit15 == 0)
  const float* pl = params + b * PPB + layer * PPL;
  cplx U[2][2];
  rot_u(pl[0], pl[1], pl[2], U);

  int i0 = b * AMPS + i;
  int i1 = i0 + HALF;
  float2 a0 = state[i0], a1 = state[i1];
  cplx s0 = { a0.x, a0.y }, s1 = { a1.x, a1.y };
  if (i & 1) { cplx t = s0; s0 = s1; s1 = t; }  // CNOT(15,0)
  cplx n0 = cadd(cmul(U[0][0], s0), cmul(U[0][1], s1));
  cplx n1 = cadd(cmul(U[1][0], s0), cmul(U[1][1], s1));
  state[i0] = make_float2(n0.re, n0.im);
  state[i1] = make_float2(n1.re, n1.im);
}

// ---------------------------------------------------------------------------
// k_local: one workgroup (1024 threads) per (batch, half) tile of 32768 amps.
//   - apply 7 fused 2-wire gates  G = [CNOT(2k,2k+1)] * (R(2k) (x) R(2k+1))
//     on wires (2,3)..(14,15)  -> bits (13,12)..(1,0)
//   - apply wire-1 gate  G1 = [X if (half && fuse)] * R(1) on bit 14
//     (X arises from CNOT(0,1): control wire0 == tile half bit)
//   - fuse_cnot==0 for the last layer; write_prob==1 writes |amp|^2 to out.
// ---------------------------------------------------------------------------
__global__ void vqc_local(const float* __restrict__ params,
                          float2* __restrict__ state,
                          float* __restrict__ out,
                          int layer, int fuse_cnot, int write_prob) {
  extern __shared__ float smem[];
  float*  tile  = smem;            // 65536 floats = 32768 complex amps
  float*  gm    = smem + 2 * HALF; // 7*64 + 16 = 464 floats of gate matrices
  float2* tile2 = (float2*)tile;

  const int tid  = threadIdx.x;
  const int b    = blockIdx.x >> 1;
  const int h    = blockIdx.x & 1;       // value of bit15 (wire 0)
  float2* gbase  = state + b * AMPS + h * HALF;

  // ---- load tile into LDS ----
#ifdef VQC_HAVE_ASYNC_LD
  {
    // 256 KB tile via async-to-LDS B128 (ASYNCcnt path, bypasses VGPRs).
    v4i* g4 = (v4i*)gbase;
    v4i* t4 = (v4i*)tile;
    #pragma unroll
    for (int k = 0; k < 16; ++k) {
      int idx = k * 1024 + tid;
      __builtin_amdgcn_global_load_async_to_lds_b128(
          (VQC_AS1 v4i*)(g4 + idx), (VQC_AS3 v4i*)(t4 + idx), 0, 0);
    }
    __builtin_amdgcn_s_wait_asynccnt(0);
  }
#else
  for (int k = 0; k < 32; ++k) {
    int idx = k * 1024 + tid;
    tile2[idx] = gbase[idx];
  }
#endif

  // ---- build gate matrices in LDS ----
  const float* pl = params + b * PPB + layer * PPL;
  if (tid < 7 * 64) {
    int j = tid >> 6, e = tid & 63;
    int w1 = 2 * j + 2, w2 = w1 + 1;
    cplx U1[2][2], U2[2][2];
    rot_u(pl[3 * w1], pl[3 * w1 + 1], pl[3 * w1 + 2], U1);
    rot_u(pl[3 * w2], pl[3 * w2 + 1], pl[3 * w2 + 2], U2);
    int r = e >> 3, c = e & 7;
    int qr = r >> 1, er = r & 1, qc = c >> 1, ec = c & 1;
    int a = qr >> 1, bb = qr & 1;
    int sb = fuse_cnot ? (bb ^ a) : bb;    // G = P_cnot * (U1 (x) U2)
    cplx g = cmul(U1[a][qc >> 1], U2[sb][qc & 1]);
    gm[j * 64 + r * 8 + c] = creal_entry(g, er, ec);
  } else if (tid < 7 * 64 + 16) {
    int e = tid - 448;
    int r = e >> 2, c = e & 3;
    cplx U1[2][2];
    rot_u(pl[3], pl[4], pl[5], U1);        // wire 1
    int qr = r >> 1, er = r & 1, qc = c >> 1, ec = c & 1;
    int sr = (fuse_cnot && h) ? (qr ^ 1) : qr;  // X * R1 on upper half
    gm[448 + r * 4 + c] = creal_entry(U1[sr][qc], er, ec);
  }
  __syncthreads();

  const int lane = tid & 31;
  const int wave = tid >> 5;
  const int sel  = lane >> 4;   // 0: K=0,1  1: K=2,3 half of the wave
  const int m    = lane & 15;   // matrix row / group column within the wave

  // ---- 7 fused 2-wire gate passes (adjacent bit pairs p2+1, p2) ----
  for (int j = 0; j < 7; ++j) {
    const int p2 = 12 - 2 * j;
    const float* M8 = gm + j * 64;
#ifdef VQC_HAVE_WMMA
    const int row = m & 7;
    const int k0  = sel << 1;
    v2f a1, a2;
    a1.x = M8[row * 8 + k0];     a1.y = M8[row * 8 + k0 + 1];
    a2.x = M8[row * 8 + 4 + k0]; a2.y = M8[row * 8 + 4 + k0 + 1];
    for (int it = wave; it < 512; it += 32) {  // 8192 groups / 16 per WMMA
      int g    = it * 16 + m;
      int i00  = ((g >> p2) << (p2 + 2)) | (g & ((1 << p2) - 1));
      int base = 2 * i00;
      int str  = 2 << p2;                      // floats between group members
      v2f b1, b2;
      int o1 = base + sel * str;               // amps 0/1 of the group
      b1.x = tile[o1]; b1.y = tile[o1 + 1];
      int o2 = base + (2 + sel) * str;         // amps 2/3 of the group
      b2.x = tile[o2]; b2.y = tile[o2 + 1];
      v8f d = {};
      d = wmma4(a1, b1, d);                    // M8[:,0:4] * X[0:4,:]
      d = wmma4(a2, b2, d);                    // + M8[:,4:8] * X[4:8,:]
      if (lane < 16) {
        #pragma unroll
        for (int r = 0; r < 8; ++r)
          tile[base + (r >> 1) * str + (r & 1)] = d[r];
      }
    }
#else
    for (int g = tid; g < 8192; g += 1024) {
      int i00  = ((g >> p2) << (p2 + 2)) | (g & ((1 << p2) - 1));
      int base = 2 * i00;
      int str  = 2 << p2;
      float x[8], y[8];
      #pragma unroll
      for (int k = 0; k < 4; ++k) {
        x[2 * k]     = tile[base + k * str];
        x[2 * k + 1] = tile[base + k * str + 1];
      }
      #pragma unroll
      for (int r = 0; r < 8; ++r) {
        float acc = 0.f;
        #pragma unroll
        for (int c = 0; c < 8; ++c) acc += M8[r * 8 + c] * x[c];
        y[r] = acc;
      }
      #pragma unroll
      for (int r = 0; r < 8; ++r)
        tile[base + (r >> 1) * str + (r & 1)] = y[r];
    }
#endif
    __syncthreads();
  }

  // ---- wire-1 gate (bit 14): 16384 pairs ----
  {
    const float* M4 = gm + 448;
#ifdef VQC_HAVE_WMMA
    const int row = m & 3;
    const int k0  = sel << 1;
    v2f a1;
    a1.x = M4[row * 4 + k0]; a1.y = M4[row * 4 + k0 + 1];
    for (int it = wave; it < 1024; it += 32) {
      int g    = it * 16 + m;
      int i0   = ((g >> 14) << 15) | (g & 0x3FFF);
      int base = 2 * i0;
      int str  = 2 << 14;
      v2f b1;
      int o = base + sel * str;
      b1.x = tile[o]; b1.y = tile[o + 1];
      v8f d = {};
      d = wmma4(a1, b1, d);
      if (lane < 16) {
        #pragma unroll
        for (int r = 0; r < 4; ++r)
          tile[base + (r >> 1) * str + (r & 1)] = d[r];
      }
    }
#else
    for (int g = tid; g < 16384; g += 1024) {
      int i0   = ((g >> 14) << 15) | (g & 0x3FFF);
      int base = 2 * i0;
      int str  = 2 << 14;
      float x[4], y[4];
      #pragma unroll
      for (int k = 0; k < 2; ++k) {
        x[2 * k]     = tile[base + k * str];
        x[2 * k + 1] = tile[base + k * str + 1];
      }
      #pragma unroll
      for (int r = 0; r < 4; ++r) {
        float acc = 0.f;
        #pragma unroll
        for (int c = 0; c < 4; ++c) acc += M4[r * 4 + c] * x[c];
        y[r] = acc;
      }
      #pragma unroll
      for (int r = 0; r < 4; ++r)
        tile[base + (r >> 1) * str + (r & 1)] = y[r];
    }
#endif
    __syncthreads();
  }

  // ---- write back: state (in-place) or |amp|^2 to d_out ----
  if (write_prob) {
    float* obase = out + b * AMPS + h * HALF;
    for (int k = 0; k < 32; ++k) {
      int idx = k * 1024 + tid;
      float re = tile[2 * idx], im = tile[2 * idx + 1];
      obase[idx] = re * re + im * im;
    }
  } else {
#ifdef VQC_HAVE_ASYNC_ST
    // 256 KB tile back to L2 via async-from-LDS B128 (ASYNCcnt path).
    v4i* g4 = (v4i*)gbase;
    v4i* t4 = (v4i*)tile;
    #pragma unroll
    for (int k = 0; k < 16; ++k) {
      int idx = k * 1024 + tid;
      __builtin_amdgcn_global_store_async_from_lds_b128(
          (VQC_AS1 v4i*)(g4 + idx), (VQC_AS3 v4i*)(t4 + idx), 0, 0);
    }
    __builtin_amdgcn_s_wait_asynccnt(0);
#else
    for (int k = 0; k < 32; ++k) {
      int idx = k * 1024 + tid;
      gbase[idx] = make_float2(tile[2 * idx], tile[2 * idx + 1]);
    }
#endif
  }
}

// ---------------------------------------------------------------------------
extern "C" void kernel_launch(void* const* d_in, const int* in_sizes, int n_in,
                              void* d_out, int out_size, void* d_ws, size_t ws_size,
                              hipStream_t stream) {
  const float* params = (const float*)d_in[0];  // [128, 480] f32
  float2* state = (float2*)d_ws;                // 128*65536 complex64 = 64 MB
  float* out = (float*)d_out;                   // 128*65536 f32

  constexpr int B = 128;
  constexpr size_t SMEM = (size_t)(2 * HALF + 512) * sizeof(float); // ~258 KB

  // |0..0> state; also equals state after layer-0's local CNOTs (all trivial).
  vqc_init<<<dim3((B * AMPS) / 1024), dim3(1024), 0, stream>>>(state);

  for (int l = 0; l < NL; ++l) {
    // CNOT(15,0) + R(0) of layer l (ordered after layer-l local CNOTs,
    // which were fused into the previous vqc_local / init).
    vqc_cross<<<dim3(HALF / 256, B), dim3(256), 0, stream>>>(params, state, l);
    const int last = (l == NL - 1);
    // R(1..15) of layer l  (+ CNOTs of layer l+1 unless last layer).
    vqc_local<<<dim3(B * 2), dim3(1024), SMEM, stream>>>(
        params, state, out, l, last ? 0 : 1, last ? 1 : 0);
  }
}